// GATBlock_MIX_20744692039831
// MI455X (gfx1250) — compile-verified
//
#include <hip/hip_runtime.h>

typedef float v2f __attribute__((ext_vector_type(2)));
typedef float v8f __attribute__((ext_vector_type(8)));

#define NNODES 50000
#define NEDGES 800000
#define NGRAPHS 500
#define NEG_SLOPE 0.2f

// ---- ordered-uint encoding for float atomicMax ----
__device__ __forceinline__ unsigned fkey(float f) {
    unsigned u = __float_as_uint(f);
    return (u & 0x80000000u) ? ~u : (u | 0x80000000u);
}
__device__ __forceinline__ float funkey(unsigned k) {
    unsigned u = (k & 0x80000000u) ? (k ^ 0x80000000u) : ~k;
    return __uint_as_float(u);
}

__device__ __forceinline__ v8f wmma_f32(v2f a, v2f b, v8f c) {
    return __builtin_amdgcn_wmma_f32_16x16x4_f32(false, a, false, b,
                                                 (short)0, c, false, false);
}

// ============================================================
// GEMM: D[N,dN] = A[N,K] @ W[K,dN]  (f32 WMMA 16x16x4)
// block = 128 threads (4 waves); wave w -> 16x32 tile
// (two 16x16 accumulators reusing one A fragment).
// Grid: (N/16, dN/128).
// ============================================================
__global__ __launch_bounds__(128)
void gemm_wmma(const float* __restrict__ A, const float* __restrict__ W,
               float* __restrict__ D, int K, int dN) {
    const int wave = threadIdx.x >> 5;
    const int lane = threadIdx.x & 31;
    const int id   = lane & 15;
    const int half = lane >> 4;
    const int rowBase = blockIdx.x * 16;
    const int col0    = blockIdx.y * 128 + wave * 32 + id;

    const float* ap = A + (size_t)(rowBase + id) * K + 2 * half;
    const float* wp = W + (size_t)(2 * half) * dN + col0;
    const size_t wstep = (size_t)4 * dN;

    v8f c0 = {}, c1 = {};
    for (int k0 = 0; k0 < K; k0 += 4) {
        v2f a;  a.x  = ap[0];      a.y  = ap[1];
        v2f b0; b0.x = wp[0];      b0.y = wp[dN];
        v2f b1; b1.x = wp[16];     b1.y = wp[dN + 16];
        c0 = wmma_f32(a, b0, c0);
        c1 = wmma_f32(a, b1, c1);
        ap += 4;
        wp += wstep;
    }
    float* dp = D + (size_t)(rowBase + half * 8) * dN + col0;
#pragma unroll
    for (int r = 0; r < 8; ++r) {
        dp[0]  = c0[r];
        dp[16] = c1[r];
        dp += dN;
    }
}

// ============================================================
// Fusion GEMM: D = concat(A1,A2)[N,2*Kh] @ W[2*Kh,dN] + bias
// Same tiling; K loop split into two segments (no per-iter select).
// ============================================================
__global__ __launch_bounds__(128)
void gemm_concat_bias(const float* __restrict__ A1, const float* __restrict__ A2,
                      const float* __restrict__ W, const float* __restrict__ bias,
                      float* __restrict__ D, int Kh, int dN) {
    const int wave = threadIdx.x >> 5;
    const int lane = threadIdx.x & 31;
    const int id   = lane & 15;
    const int half = lane >> 4;
    const int rowBase = blockIdx.x * 16;
    const int col0    = blockIdx.y * 128 + wave * 32 + id;

    const float* wp = W + (size_t)(2 * half) * dN + col0;
    const size_t wstep = (size_t)4 * dN;

    v8f c0 = {}, c1 = {};
    const float* segs[2] = {A1, A2};
#pragma unroll
    for (int seg = 0; seg < 2; ++seg) {
        const float* ap = segs[seg] + (size_t)(rowBase + id) * Kh + 2 * half;
        for (int k0 = 0; k0 < Kh; k0 += 4) {
            v2f a;  a.x  = ap[0];      a.y  = ap[1];
            v2f b0; b0.x = wp[0];      b0.y = wp[dN];
            v2f b1; b1.x = wp[16];     b1.y = wp[dN + 16];
            c0 = wmma_f32(a, b0, c0);
            c1 = wmma_f32(a, b1, c1);
            ap += 4;
            wp += wstep;
        }
    }
    const float bc0 = bias[col0];
    const float bc1 = bias[col0 + 16];
    float* dp = D + (size_t)(rowBase + half * 8) * dN + col0;
#pragma unroll
    for (int r = 0; r < 8; ++r) {
        dp[0]  = c0[r] + bc0;
        dp[16] = c1[r] + bc1;
        dp += dN;
    }
}

// ============================================================
// Per-node attention scores: as = h.att_s, ad = h.att_d  (wave/node)
// ============================================================
__global__ void scores_kernel(const float* __restrict__ h,
                              const float* __restrict__ att_s,
                              const float* __restrict__ att_d,
                              float* __restrict__ as_, float* __restrict__ ad_,
                              int dN) {
    const int gw   = (blockIdx.x * blockDim.x + threadIdx.x) >> 5;
    const int lane = threadIdx.x & 31;
    if (gw >= NNODES) return;
    const float* hp = h + (size_t)gw * dN;
    float ss = 0.f, sd = 0.f;
    for (int k = lane; k < dN; k += 32) {
        const float v = hp[k];
        ss += v * att_s[k];
        sd += v * att_d[k];
    }
#pragma unroll
    for (int off = 16; off; off >>= 1) {
        ss += __shfl_xor(ss, off, 32);
        sd += __shfl_xor(sd, off, 32);
    }
    if (lane == 0) { as_[gw] = ss; ad_[gw] = sd; }
}

// zero aggregation target + per-node softmax state
__global__ void init_agg(float* __restrict__ agg, unsigned* __restrict__ mkey,
                         float* __restrict__ den, int total, int n) {
    const int i = blockIdx.x * blockDim.x + threadIdx.x;
    if (i < total) agg[i] = 0.f;
    if (i < n) { mkey[i] = 0u; den[i] = 0.f; }
}

// pass 1: e = leaky_relu(as[s]+ad[d]); segment max via uint atomicMax
__global__ void edge_max(const int* __restrict__ src, const int* __restrict__ dst,
                         const float* __restrict__ as_, const float* __restrict__ ad_,
                         unsigned* __restrict__ mkey, float* __restrict__ ebuf,
                         int E, int n) {
    const int i = blockIdx.x * blockDim.x + threadIdx.x;
    if (i >= E + n) return;
    int s, d;
    if (i < E) { s = src[i]; d = dst[i]; } else { s = d = i - E; }
    float e = as_[s] + ad_[d];
    e = e > 0.f ? e : NEG_SLOPE * e;
    ebuf[i] = e;
    atomicMax(&mkey[d], fkey(e));
}

// pass 2: ex = exp(e - m[d]); den[d] += ex
__global__ void edge_exp(const int* __restrict__ dst,
                         const unsigned* __restrict__ mkey,
                         float* __restrict__ den, float* __restrict__ ebuf,
                         int E, int n) {
    const int i = blockIdx.x * blockDim.x + threadIdx.x;
    if (i >= E + n) return;
    const int d = (i < E) ? dst[i] : i - E;
    const float ex = expf(ebuf[i] - funkey(mkey[d]));
    ebuf[i] = ex;
    unsafeAtomicAdd(&den[d], ex);
}

// pass 3: agg[d,:] += (ex/den[d]) * h[s,:]   (wave/edge, lane-striped dims)
__global__ void edge_aggregate(const int* __restrict__ src, const int* __restrict__ dst,
                               const float* __restrict__ ebuf, const float* __restrict__ den,
                               const float* __restrict__ h, float* __restrict__ agg,
                               int E, int n, int dN) {
    const int gw   = (blockIdx.x * blockDim.x + threadIdx.x) >> 5;
    const int lane = threadIdx.x & 31;
    if (gw >= E + n) return;
    int s, d;
    if (gw < E) { s = src[gw]; d = dst[gw]; } else { s = d = gw - E; }
    const float alpha = ebuf[gw] / den[d];
    const float* hp = h + (size_t)s * dN;
    float* op = agg + (size_t)d * dN;
    for (int k = lane; k < dN; k += 32)
        unsafeAtomicAdd(&op[k], alpha * hp[k]);
}

__global__ void bias_act(float* __restrict__ agg, const float* __restrict__ bias,
                         int dN, int total, int do_relu) {
    const int i = blockIdx.x * blockDim.x + threadIdx.x;
    if (i >= total) return;
    float v = agg[i] + bias[i & (dN - 1)];
    if (do_relu) v = fmaxf(v, 0.f);
    agg[i] = v;
}

// ---- global mean pool over batch ----
__global__ void pool_init(float* __restrict__ num, float* __restrict__ cnt, int dN) {
    const int i = blockIdx.x * blockDim.x + threadIdx.x;
    if (i < NGRAPHS * dN) num[i] = 0.f;
    if (i < NGRAPHS) cnt[i] = 0.f;
}
__global__ void pool_sum(const float* __restrict__ out, const int* __restrict__ batch,
                         float* __restrict__ num, float* __restrict__ cnt, int dN) {
    const int gw   = (blockIdx.x * blockDim.x + threadIdx.x) >> 5;
    const int lane = threadIdx.x & 31;
    if (gw >= NNODES) return;
    const int g = batch[gw];
    if (lane == 0) unsafeAtomicAdd(&cnt[g], 1.f);
    const float* op = out + (size_t)gw * dN;
    float* np = num + (size_t)g * dN;
    for (int k = lane; k < dN; k += 32)
        unsafeAtomicAdd(&np[k], op[k]);
}
__global__ void pool_div(const float* __restrict__ num, const float* __restrict__ cnt,
                         float* __restrict__ out, int dN) {
    const int i = blockIdx.x * blockDim.x + threadIdx.x;
    if (i >= NGRAPHS * dN) return;
    out[i] = num[i] / fmaxf(cnt[i / dN], 1.f);
}

extern "C" void kernel_launch(void* const* d_in, const int* in_sizes, int n_in,
                              void* d_out, int out_size, void* d_ws, size_t ws_size,
                              hipStream_t stream) {
    const float* x   = (const float*)d_in[0];
    const int* ei    = (const int*)d_in[1];
    const int* batch = (const int*)d_in[3];
    const int* ein   = (const int*)d_in[4];
    const int* src_o = ei,  *dst_o = ei  + NEDGES;
    const int* src_n = ein, *dst_n = ein + NEDGES;
    // params start at index 7; 10 arrays per layer:
    // Wo, aso, ado, bo, Wn, asn, adn, bn, Wf, bf
    auto P = [&](int layer, int j) { return (const float*)d_in[7 + 10 * layer + j]; };

    // ---- workspace layout (floats) ----
    float* ws = (float*)d_ws;
    const size_t NC = (size_t)NNODES * 256;
    float* bufA_o = ws;
    float* bufA_n = bufA_o + NC;
    float* bufB_o = bufA_n + NC;
    float* bufB_n = bufB_o + NC;
    float* hbuf   = bufB_n + NC;
    float* fused  = hbuf + NC;
    float* as_    = fused + NC;
    float* ad_    = as_ + NNODES;
    unsigned* mkey = (unsigned*)(ad_ + NNODES);
    float* den    = (float*)(mkey + NNODES);
    float* ebuf   = den + NNODES;
    float* num    = ebuf + (NEDGES + NNODES);
    float* cnt    = num + (size_t)NGRAPHS * 256;

    const int EN = NEDGES + NNODES;
    const int enBlocks   = (EN + 255) / 256;
    const int waveNodeBk = (NNODES * 32 + 255) / 256;
    const unsigned waveEdgeBk = (unsigned)(((size_t)EN * 32 + 255) / 256);

    auto run_branch = [&](const float* in, float* out, const int* src, const int* dst,
                          const float* W, const float* atts, const float* attd,
                          const float* bias, int dIn, int dOut, int relu) {
        dim3 gg(NNODES / 16, dOut / 128);
        gemm_wmma<<<gg, 128, 0, stream>>>(in, W, hbuf, dIn, dOut);
        scores_kernel<<<waveNodeBk, 256, 0, stream>>>(hbuf, atts, attd, as_, ad_, dOut);
        const int total = NNODES * dOut;
        init_agg<<<(total + 255) / 256, 256, 0, stream>>>(out, mkey, den, total, NNODES);
        edge_max<<<enBlocks, 256, 0, stream>>>(src, dst, as_, ad_, mkey, ebuf, NEDGES, NNODES);
        edge_exp<<<enBlocks, 256, 0, stream>>>(dst, mkey, den, ebuf, NEDGES, NNODES);
        edge_aggregate<<<waveEdgeBk, 256, 0, stream>>>(src, dst, ebuf, den, hbuf, out,
                                                       NEDGES, NNODES, dOut);
        bias_act<<<(total + 255) / 256, 256, 0, stream>>>(out, bias, dOut, total, relu);
    };

    const int dims[4] = {128, 128, 128, 256};
    const float* in_o = x;
    const float* in_n = x;
    float* po = bufA_o; float* pn = bufA_n;
    float* qo = bufB_o; float* qn = bufB_n;

    for (int i = 0; i < 3; ++i) {
        const int dIn = dims[i], dOut = dims[i + 1];
        run_branch(in_o, po, src_o, dst_o, P(i, 0), P(i, 1), P(i, 2), P(i, 3), dIn, dOut, i > 0);
        run_branch(in_n, pn, src_n, dst_n, P(i, 4), P(i, 5), P(i, 6), P(i, 7), dIn, dOut, i > 0);
        in_o = po; in_n = pn;
        float* t;
        t = po; po = qo; qo = t;
        t = pn; pn = qn; qn = t;
    }

    // only the final fusion (i=2) matters: out = concat(out_o,out_n) @ Wf2 + bf2
    dim3 gf(NNODES / 16, 256 / 128);
    gemm_concat_bias<<<gf, 128, 0, stream>>>(in_o, in_n, P(2, 8), P(2, 9), fused, 256, 256);

    // global mean pool over `batch`
    pool_init<<<(NGRAPHS * 256 + 255) / 256, 256, 0, stream>>>(num, cnt, 256);
    pool_sum<<<waveNodeBk, 256, 0, stream>>>(fused, batch, num, cnt, 256);
    pool_div<<<(NGRAPHS * 256 + 255) / 256, 256, 0, stream>>>(num, cnt, (float*)d_out, 256);
}